// TRNN_64364379898292
// MI455X (gfx1250) — compile-verified
//
#include <hip/hip_runtime.h>

typedef __attribute__((ext_vector_type(2))) float v2f;
typedef __attribute__((ext_vector_type(8))) float v8f;

#define HIDDEN 20
#define SEQ 50
#define GATES 80     // 4*HIDDEN
#define WAVES 8
#define TILE_B 16

__device__ __forceinline__ float fast_exp(float x) {
    return __builtin_amdgcn_exp2f(x * 1.4426950408889634f);
}
__device__ __forceinline__ float fast_sigmoid(float x) {
    return __builtin_amdgcn_rcpf(1.0f + fast_exp(-x));
}
__device__ __forceinline__ float fast_tanh(float x) {
    return 1.0f - 2.0f * __builtin_amdgcn_rcpf(1.0f + fast_exp(2.0f * x));
}

__global__ __launch_bounds__(256) void lstm_wmma_kernel(
    const float* __restrict__ x, const float* __restrict__ w_ih,
    const float* __restrict__ w_hh, const float* __restrict__ b_ih,
    const float* __restrict__ b_hh, const float* __restrict__ fc_w,
    const float* __restrict__ fc_b, float* __restrict__ out, int B)
{
    __shared__ float fcbuf[2 * SEQ * HIDDEN];        // 2000 floats, block-shared
    __shared__ float xbuf[WAVES][SEQ * TILE_B];      // [wave][t*16 + m]
    __shared__ float gbuf[WAVES][GATES * TILE_B];    // [wave][n*16 + m]
    __shared__ float hbuf[WAVES][TILE_B * HIDDEN];   // [wave][m*20 + h]

    const int tid  = threadIdx.x;
    const int wid  = tid >> 5;    // wave32
    const int lane = tid & 31;
    const int half = lane >> 4;   // 0: lanes 0-15, 1: lanes 16-31
    const int nl   = lane & 15;

    const int b0 = (blockIdx.x * WAVES + wid) * TILE_B;
    const bool active = (b0 + TILE_B) <= B;

    // stage FC weights once per block
    for (int i = tid; i < 2 * SEQ * HIDDEN; i += blockDim.x) fcbuf[i] = fc_w[i];

    if (active) {
        // stage x tile for this wave as [t][m]
        for (int i = lane; i < SEQ * TILE_B; i += 32) {
            int t = i >> 4, m = i & 15;
            xbuf[wid][t * TILE_B + m] = x[(size_t)(b0 + m) * SEQ + t];
        }
        // zero initial hidden state
        for (int i = lane; i < TILE_B * HIDDEN; i += 32) hbuf[wid][i] = 0.0f;
    }
    __syncthreads();
    if (!active) return;

    // B-fragments of w_hh^T: B[k][n] = w_hh[n][k]
    // lane layout (32-bit B, 4x16): lanes 0-15 hold K = 4kt+0,1 ; lanes 16-31 hold K = 4kt+2,3
    v2f bfrag[5][5];
    #pragma unroll
    for (int kt = 0; kt < 5; ++kt)
        #pragma unroll
        for (int nt = 0; nt < 5; ++nt) {
            int row = nt * 16 + nl;        // gate output index n
            int k0  = 4 * kt + 2 * half;   // hidden index k
            bfrag[kt][nt].x = w_hh[row * HIDDEN + k0];
            bfrag[kt][nt].y = w_hh[row * HIDDEN + k0 + 1];
        }

    // per-lane input-projection weight/bias for column n = nt*16 + nl
    float wn[5], bn[5];
    #pragma unroll
    for (int nt = 0; nt < 5; ++nt) {
        int col = nt * 16 + nl;
        wn[nt] = w_ih[col];
        bn[nt] = b_ih[col] + b_hh[col];
    }

    const int moff = half * 8;        // D-layout row offset: VGPR r holds M = r + moff
    const int bown = lane >> 1;       // batch row this lane owns in gate math
    const int hb   = (lane & 1) * 10; // hidden base this lane owns

    float cst[10];
    #pragma unroll
    for (int j = 0; j < 10; ++j) cst[j] = 0.0f;
    float facc0 = 0.0f, facc1 = 0.0f;

    #pragma unroll 1
    for (int t = 0; t < SEQ; ++t) {
        // A-fragments of h (16x20): lane L<16: M=L, K=4kt+0,1 ; lane L>=16: M=L-16, K=4kt+2,3
        v2f afrag[5];
        #pragma unroll
        for (int kt = 0; kt < 5; ++kt) {
            int k0 = 4 * kt + 2 * half;
            afrag[kt].x = hbuf[wid][nl * HIDDEN + k0];
            afrag[kt].y = hbuf[wid][nl * HIDDEN + k0 + 1];
        }
        float xm[8];
        #pragma unroll
        for (int r = 0; r < 8; ++r) xm[r] = xbuf[wid][t * TILE_B + moff + r];

        // gates = xp + h @ w_hh^T, via 5x5 f32 WMMAs
        #pragma unroll
        for (int nt = 0; nt < 5; ++nt) {
            v8f acc;
            #pragma unroll
            for (int r = 0; r < 8; ++r) acc[r] = fmaf(xm[r], wn[nt], bn[nt]);
            #pragma unroll
            for (int kt = 0; kt < 5; ++kt)
                acc = __builtin_amdgcn_wmma_f32_16x16x4_f32(
                    false, afrag[kt], false, bfrag[kt][nt],
                    (short)0, acc, false, false);
            #pragma unroll
            for (int r = 0; r < 8; ++r)
                gbuf[wid][(nt * 16 + nl) * TILE_B + moff + r] = acc[r];
        }

        // per-wave LDS exchange: DS ops are in-order within a wave; fence the compiler
        asm volatile("" ::: "memory");

        // gate nonlinearities + state update; lane owns (bown, hb..hb+9)
        #pragma unroll
        for (int j = 0; j < 10; ++j) {
            int h = hb + j;
            float iv = fast_sigmoid(gbuf[wid][(h             ) * TILE_B + bown]);
            float fv = fast_sigmoid(gbuf[wid][(HIDDEN     + h) * TILE_B + bown]);
            float gv = fast_tanh   (gbuf[wid][(2 * HIDDEN + h) * TILE_B + bown]);
            float ov = fast_sigmoid(gbuf[wid][(3 * HIDDEN + h) * TILE_B + bown]);
            float c  = fv * cst[j] + iv * gv;
            cst[j]   = c;
            float hv = ov * fast_tanh(c);
            hbuf[wid][bown * HIDDEN + h] = hv;
            // fused FC accumulation: flat index = t*HIDDEN + h
            facc0 = fmaf(hv, fcbuf[t * HIDDEN + h], facc0);
            facc1 = fmaf(hv, fcbuf[SEQ * HIDDEN + t * HIDDEN + h], facc1);
        }
        asm volatile("" ::: "memory");
    }

    // combine the two lanes owning batch row bown (h0..9 and h10..19)
    facc0 += __shfl_xor(facc0, 1, 32);
    facc1 += __shfl_xor(facc1, 1, 32);
    if ((lane & 1) == 0) {
        out[(size_t)(b0 + bown) * 2 + 0] = facc0 + fc_b[0];
        out[(size_t)(b0 + bown) * 2 + 1] = facc1 + fc_b[1];
    }
}

extern "C" void kernel_launch(void* const* d_in, const int* in_sizes, int n_in,
                              void* d_out, int out_size, void* d_ws, size_t ws_size,
                              hipStream_t stream) {
    const float* x    = (const float*)d_in[0];
    const float* w_ih = (const float*)d_in[1];
    const float* w_hh = (const float*)d_in[2];
    const float* b_ih = (const float*)d_in[3];
    const float* b_hh = (const float*)d_in[4];
    const float* fc_w = (const float*)d_in[5];
    const float* fc_b = (const float*)d_in[6];
    float* out = (float*)d_out;

    int B = in_sizes[0] / SEQ;                 // x has B*SEQ elements
    int perBlock = WAVES * TILE_B;             // 128 batches per block
    int grid = (B + perBlock - 1) / perBlock;  // 512 for B=65536

    lstm_wmma_kernel<<<grid, 256, 0, stream>>>(x, w_ih, w_hh, b_ih, b_hh,
                                               fc_w, fc_b, out, B);
}